// AALModel_317827579936
// MI455X (gfx1250) — compile-verified
//
#include <hip/hip_runtime.h>
#include <hip/hip_bf16.h>
#include <math.h>

#define NUM_GRAPHS 4096
#define NPG 116
#define NNODES (NUM_GRAPHS * NPG)   // 475136
#define NEDGES (NNODES * 28)        // 13303808

typedef float v2f __attribute__((ext_vector_type(2)));
typedef float v8f __attribute__((ext_vector_type(8)));

// ---------------------------------------------------------------- utilities
__global__ void zero_kernel(float* __restrict__ p, int n) {
  int i = blockIdx.x * blockDim.x + threadIdx.x;
  if (i < n) p[i] = 0.0f;
}

// recompute ew1 = relu([x0_s, x0_d, ew0] @ dom1_W + dom1_b)   (3 -> 2)
__device__ __forceinline__ void ew1_of(float xs, float xd, float w0,
                                       const float* __restrict__ d1W,
                                       const float* __restrict__ d1b,
                                       float& e10, float& e11) {
  e10 = fmaxf(0.0f, xs * d1W[0] + xd * d1W[2] + w0 * d1W[4] + d1b[0]);
  e11 = fmaxf(0.0f, xs * d1W[1] + xd * d1W[3] + w0 * d1W[5] + d1b[1]);
}

// ---------------------------------------------------------------- pass 1
__global__ void edge_pass1(const int* __restrict__ ei,
                           const float* __restrict__ x0,
                           const float* __restrict__ ea,
                           const float* __restrict__ d1W, const float* __restrict__ d1b,
                           const float* __restrict__ n1W, const float* __restrict__ n1b,
                           float* __restrict__ agg1,
                           float* __restrict__ ewsum,
                           float* __restrict__ ecount) {
  int e = blockIdx.x * blockDim.x + threadIdx.x;
  if (e >= NEDGES) return;
  int s = ei[e], d = ei[NEDGES + e];
  float xs = x0[s], xd = x0[d], w0 = ea[e];
  float e10, e11;
  ew1_of(xs, xd, w0, d1W, d1b, e10, e11);
  float w1 = fmaxf(0.0f, e10 * n1W[0] + e11 * n1W[1] + n1b[0]);
  unsafeAtomicAdd(&agg1[d], w1 * xs);
  int gb = s / NPG;
  float* es = ewsum + gb * 10;
  unsafeAtomicAdd(es + 0, w0);
  unsafeAtomicAdd(es + 1, e10);
  unsafeAtomicAdd(es + 2, e11);
  unsafeAtomicAdd(&ecount[gb], 1.0f);
}

__global__ void node_pass1(const float* __restrict__ x0,
                           const float* __restrict__ agg1,
                           const float* __restrict__ Wrel, const float* __restrict__ b,
                           const float* __restrict__ Wroot,
                           float* __restrict__ x1) {
  int n = blockIdx.x * blockDim.x + threadIdx.x;
  if (n >= NNODES) return;
  float a = agg1[n], x = x0[n];
#pragma unroll
  for (int j = 0; j < 3; ++j)
    x1[n * 3 + j] = fmaxf(0.0f, a * Wrel[j] + b[j] + x * Wroot[j]);
}

// ---------------------------------------------------------------- pass 2
__global__ void edge_pass2(const int* __restrict__ ei,
                           const float* __restrict__ x0,
                           const float* __restrict__ ea,
                           const float* __restrict__ x1,
                           const float* __restrict__ d1W, const float* __restrict__ d1b,
                           const float* __restrict__ d2W, const float* __restrict__ d2b,
                           const float* __restrict__ n2W, const float* __restrict__ n2b,
                           float* __restrict__ agg2,
                           float* __restrict__ ewsum) {
  int e = blockIdx.x * blockDim.x + threadIdx.x;
  if (e >= NEDGES) return;
  int s = ei[e], d = ei[NEDGES + e];
  float e10, e11;
  ew1_of(x0[s], x0[d], ea[e], d1W, d1b, e10, e11);
  float f[8];
#pragma unroll
  for (int i = 0; i < 3; ++i) { f[i] = x1[s * 3 + i]; f[3 + i] = x1[d * 3 + i]; }
  f[6] = e10; f[7] = e11;
  float ew2[3];
#pragma unroll
  for (int j = 0; j < 3; ++j) {
    float acc = d2b[j];
#pragma unroll
    for (int i = 0; i < 8; ++i) acc += f[i] * d2W[i * 3 + j];
    ew2[j] = fmaxf(0.0f, acc);
  }
  float w2 = fmaxf(0.0f, ew2[0] * n2W[0] + ew2[1] * n2W[1] + ew2[2] * n2W[2] + n2b[0]);
#pragma unroll
  for (int i = 0; i < 3; ++i) unsafeAtomicAdd(&agg2[d * 3 + i], w2 * f[i]);
  int gb = s / NPG;
  float* es = ewsum + gb * 10 + 3;
#pragma unroll
  for (int j = 0; j < 3; ++j) unsafeAtomicAdd(es + j, ew2[j]);
}

__global__ void node_pass23(const float* __restrict__ xin,   // [N,3]
                            const float* __restrict__ agg,   // [N,3]
                            const float* __restrict__ Wrel,  // [3,OUT]
                            const float* __restrict__ b,
                            const float* __restrict__ Wroot, // [3,OUT]
                            float* __restrict__ xout, int OUT) {
  int n = blockIdx.x * blockDim.x + threadIdx.x;
  if (n >= NNODES) return;
  float a0 = agg[n * 3], a1 = agg[n * 3 + 1], a2 = agg[n * 3 + 2];
  float y0 = xin[n * 3], y1 = xin[n * 3 + 1], y2 = xin[n * 3 + 2];
  for (int j = 0; j < OUT; ++j) {
    float acc = b[j] + a0 * Wrel[j] + a1 * Wrel[OUT + j] + a2 * Wrel[2 * OUT + j]
                    + y0 * Wroot[j] + y1 * Wroot[OUT + j] + y2 * Wroot[2 * OUT + j];
    xout[n * OUT + j] = fmaxf(0.0f, acc);
  }
}

// ---------------------------------------------------------------- pass 3
__global__ void edge_pass3(const int* __restrict__ ei,
                           const float* __restrict__ x0,
                           const float* __restrict__ ea,
                           const float* __restrict__ x1,
                           const float* __restrict__ x2,
                           const float* __restrict__ d1W, const float* __restrict__ d1b,
                           const float* __restrict__ d2W, const float* __restrict__ d2b,
                           const float* __restrict__ d3W, const float* __restrict__ d3b,
                           const float* __restrict__ n3W, const float* __restrict__ n3b,
                           float* __restrict__ agg3,
                           float* __restrict__ ewsum) {
  int e = blockIdx.x * blockDim.x + threadIdx.x;
  if (e >= NEDGES) return;
  int s = ei[e], d = ei[NEDGES + e];
  // recompute ew1
  float e10, e11;
  ew1_of(x0[s], x0[d], ea[e], d1W, d1b, e10, e11);
  // recompute ew2
  float f8[8];
#pragma unroll
  for (int i = 0; i < 3; ++i) { f8[i] = x1[s * 3 + i]; f8[3 + i] = x1[d * 3 + i]; }
  f8[6] = e10; f8[7] = e11;
  float ew2[3];
#pragma unroll
  for (int j = 0; j < 3; ++j) {
    float acc = d2b[j];
#pragma unroll
    for (int i = 0; i < 8; ++i) acc += f8[i] * d2W[i * 3 + j];
    ew2[j] = fmaxf(0.0f, acc);
  }
  // ew3 (no relu)
  float f9[9];
#pragma unroll
  for (int i = 0; i < 3; ++i) { f9[i] = x2[s * 3 + i]; f9[3 + i] = x2[d * 3 + i]; f9[6 + i] = ew2[i]; }
  float ew3[4];
#pragma unroll
  for (int j = 0; j < 4; ++j) {
    float acc = d3b[j];
#pragma unroll
    for (int i = 0; i < 9; ++i) acc += f9[i] * d3W[i * 4 + j];
    ew3[j] = acc;
  }
  float w3 = n3b[0];
#pragma unroll
  for (int j = 0; j < 4; ++j) w3 += ew3[j] * n3W[j];
  w3 = fmaxf(0.0f, w3);
#pragma unroll
  for (int i = 0; i < 3; ++i) unsafeAtomicAdd(&agg3[d * 3 + i], w3 * f9[i]);
  int gb = s / NPG;
  float* es = ewsum + gb * 10 + 6;
#pragma unroll
  for (int j = 0; j < 4; ++j) unsafeAtomicAdd(es + j, ew3[j]);
}

// ---------------------------------------------------------------- pooling
// one wave per graph; nodes of graph b are contiguous [b*116, b*116+116)
// writes pooled with stride 24 (element 23 = 0 pad for the WMMA K dimension)
__global__ void pool_kernel(const float* __restrict__ x0,
                            const float* __restrict__ x1,
                            const float* __restrict__ x2,
                            const float* __restrict__ x3,
                            const float* __restrict__ ewsum,
                            const float* __restrict__ ecount,
                            const float* __restrict__ g,
                            float* __restrict__ pooled) {
  int b = blockIdx.x;
  int t = threadIdx.x;
  int base = b * NPG;
  if (t < 12) {
    float s = 0.0f;
    for (int i = 0; i < NPG; ++i) {
      int n = base + i;
      float v;
      if (t == 0)      v = x0[n];
      else if (t < 4)  v = x1[n * 3 + (t - 1)];
      else if (t < 7)  v = x2[n * 3 + (t - 4)];
      else             v = x3[n * 5 + (t - 7)];
      s += v;
    }
    pooled[b * 24 + t] = s * (1.0f / (float)NPG);
  } else if (t < 22) {
    float c = fmaxf(ecount[b], 1.0f);
    pooled[b * 24 + t] = ewsum[b * 10 + (t - 12)] / c;
  } else if (t == 22) {
    pooled[b * 24 + 22] = g[b];
  } else if (t == 23) {
    pooled[b * 24 + 23] = 0.0f;    // K padding
  }
}

// ---------------------------------------------------------------- classifier
// pooled[4096,24(pad)] @ mlp_W[23,2] + b, log_softmax.  One wave -> 16 graphs
// (M), K=24 as six V_WMMA_F32_16X16X4_F32 steps, N padded 2->16.  B matrix is
// staged zero-padded in LDS so the inner loop is unconditional loads + wmma.
__global__ void classifier_wmma(const float* __restrict__ pooled,
                                const float* __restrict__ mlpW,
                                const float* __restrict__ mlpb,
                                float* __restrict__ out) {
  __shared__ float sB[24][16];   // B[k][n], zero-padded (rows>=23, cols>=2)
  __shared__ float ls[8][16][2];

  // stage padded B once per block (384 entries, 256 threads)
  for (int i = threadIdx.x; i < 24 * 16; i += 256) {
    int k = i >> 4, n = i & 15;
    sB[k][n] = (k < 23 && n < 2) ? mlpW[k * 2 + n] : 0.0f;
  }
  __syncthreads();

  int lane = threadIdx.x & 31;
  int wid  = threadIdx.x >> 5;
  int g0   = (blockIdx.x * 8 + wid) * 16;

  int m  = lane & 15;          // M row (A) / N col (B)
  int kk = (lane >> 4) * 2;    // lanes 0-15: K=0,1 ; lanes 16-31: K=2,3
  const float* arow = pooled + (size_t)(g0 + m) * 24 + kk;

  v8f c = {};
#pragma unroll
  for (int k0 = 0; k0 < 24; k0 += 4) {
    v2f a, bb;
    a.x  = arow[k0];
    a.y  = arow[k0 + 1];
    bb.x = sB[k0 + kk][m];
    bb.y = sB[k0 + kk + 1][m];
    c = __builtin_amdgcn_wmma_f32_16x16x4_f32(false, a, false, bb,
                                              (short)0, c, false, false);
  }

  // D layout: lanes 0-15 hold rows M=r (VGPR r), lanes 16-31 hold M=r+8, N=lane&15
  if (m < 2) {
    int mbase = (lane >= 16) ? 8 : 0;
#pragma unroll
    for (int r = 0; r < 8; ++r) ls[wid][mbase + r][m] = c[r] + mlpb[m];
  }
  __syncthreads();
  if (lane < 16) {
    float l0 = ls[wid][lane][0], l1 = ls[wid][lane][1];
    float mx = fmaxf(l0, l1);
    float lse = mx + logf(expf(l0 - mx) + expf(l1 - mx));
    out[(g0 + lane) * 2 + 0] = l0 - lse;
    out[(g0 + lane) * 2 + 1] = l1 - lse;
  }
}

// ---------------------------------------------------------------- launch
extern "C" void kernel_launch(void* const* d_in, const int* in_sizes, int n_in,
                              void* d_out, int out_size, void* d_ws, size_t ws_size,
                              hipStream_t stream) {
  const float* x0  = (const float*)d_in[0];
  const int*   ei  = (const int*)d_in[1];
  const float* ea  = (const float*)d_in[2];
  const float* g   = (const float*)d_in[3];
  // d_in[4] = batch (unused; graphs are contiguous blocks of 116 nodes)
  const float* c1Wrel = (const float*)d_in[5];
  const float* c1b    = (const float*)d_in[6];
  const float* c1Wroot= (const float*)d_in[7];
  const float* c2Wrel = (const float*)d_in[8];
  const float* c2b    = (const float*)d_in[9];
  const float* c2Wroot= (const float*)d_in[10];
  const float* c3Wrel = (const float*)d_in[11];
  const float* c3b    = (const float*)d_in[12];
  const float* c3Wroot= (const float*)d_in[13];
  const float* d1W = (const float*)d_in[14];
  const float* d1b = (const float*)d_in[15];
  const float* d2W = (const float*)d_in[16];
  const float* d2b = (const float*)d_in[17];
  const float* d3W = (const float*)d_in[18];
  const float* d3b = (const float*)d_in[19];
  const float* n1W = (const float*)d_in[20];
  const float* n1b = (const float*)d_in[21];
  const float* n2W = (const float*)d_in[22];
  const float* n2b = (const float*)d_in[23];
  const float* n3W = (const float*)d_in[24];
  const float* n3b = (const float*)d_in[25];
  const float* mlpW = (const float*)d_in[26];
  const float* mlpb = (const float*)d_in[27];
  float* out = (float*)d_out;

  // workspace layout (floats)
  float* ws     = (float*)d_ws;
  float* agg1   = ws;                       // N
  float* agg2   = agg1 + NNODES;            // 3N
  float* agg3   = agg2 + 3 * NNODES;        // 3N
  float* ewsum  = agg3 + 3 * NNODES;        // 10*B
  float* ecount = ewsum + 10 * NUM_GRAPHS;  // B
  float* x1     = ecount + NUM_GRAPHS;      // 3N
  float* x2     = x1 + 3 * NNODES;          // 3N
  float* x3     = x2 + 3 * NNODES;          // 5N
  float* pooled = x3 + 5 * NNODES;          // 24*B (stride-24 padded)

  const int nzero = 7 * NNODES + 11 * NUM_GRAPHS;
  zero_kernel<<<(nzero + 255) / 256, 256, 0, stream>>>(ws, nzero);

  const int EB = NEDGES / 256;   // 51968, exact
  const int NB = NNODES / 256;   // 1856, exact

  edge_pass1<<<EB, 256, 0, stream>>>(ei, x0, ea, d1W, d1b, n1W, n1b,
                                     agg1, ewsum, ecount);
  node_pass1<<<NB, 256, 0, stream>>>(x0, agg1, c1Wrel, c1b, c1Wroot, x1);

  edge_pass2<<<EB, 256, 0, stream>>>(ei, x0, ea, x1, d1W, d1b, d2W, d2b,
                                     n2W, n2b, agg2, ewsum);
  node_pass23<<<NB, 256, 0, stream>>>(x1, agg2, c2Wrel, c2b, c2Wroot, x2, 3);

  edge_pass3<<<EB, 256, 0, stream>>>(ei, x0, ea, x1, x2, d1W, d1b, d2W, d2b,
                                     d3W, d3b, n3W, n3b, agg3, ewsum);
  node_pass23<<<NB, 256, 0, stream>>>(x2, agg3, c3Wrel, c3b, c3Wroot, x3, 5);

  pool_kernel<<<NUM_GRAPHS, 32, 0, stream>>>(x0, x1, x2, x3, ewsum, ecount, g, pooled);

  // 4096 graphs = 32 blocks * 8 waves * 16 graphs (exact -> EXEC all-ones for WMMA)
  classifier_wmma<<<NUM_GRAPHS / 128, 256, 0, stream>>>(pooled, mlpW, mlpb, out);
}